// RayleighBenardFullPDELoss_5119601017344
// MI455X (gfx1250) — compile-verified
//
#include <hip/hip_runtime.h>

// ---- problem constants (from the reference) ----
#define NXc   512
#define NYc   128
#define Bc    8
#define Tc    32
#define TINT  30          // interior time slices (T-2)
#define SKIPc 15
#define XPB   8           // x-points per block

typedef float v2f __attribute__((ext_vector_type(2)));
typedef float v8f __attribute__((ext_vector_type(8)));

// Wave32 sum via V_WMMA_F32_16X16X4_F32 with an all-ones B matrix.
// A (16x4 f32): lane L in [0,15] holds K=0,1 in a.x/a.y ; lanes 16..31 hold K=2,3.
// D[m][n] = sum_k A[m][k]  ->  row m sums lanes {m, m+16} (both A VGPRs).
// Per-lane sum of D's 8 VGPRs covers rows 0..7 (lanes 0-15) or 8..15 (lanes 16-31);
// one shfl_xor(16) completes the full 32-lane sum in every lane.
__device__ __forceinline__ float wave_reduce_wmma(float s) {
    v2f a;  a.x = s;    a.y = 0.0f;
    v2f bm; bm.x = 1.0f; bm.y = 1.0f;
    v8f c = {};
    c = __builtin_amdgcn_wmma_f32_16x16x4_f32(
            /*neg_a=*/false, a, /*neg_b=*/false, bm,
            /*c_mod=*/(short)0, c, /*reuse_a=*/false, /*reuse_b=*/false);
    float p = c[0] + c[1] + c[2] + c[3] + c[4] + c[5] + c[6] + c[7];
    p += __shfl_xor(p, 16, 32);
    return p;
}

__global__ void __launch_bounds__(128)
rb_loss_kernel(const float* __restrict__ u, const float* __restrict__ v,
               const float* __restrict__ bb_, const float* __restrict__ kappa,
               float* __restrict__ partials)
{
    const int y  = threadIdx.x;          // 0..127, contiguous axis -> coalesced
    const int bt = blockIdx.y;           // 0..239
    const int ib = bt / TINT;            // batch
    const int ti = (bt % TINT) + 1;      // interior time index 1..30

    const float kap = kappa[ib];

    const long long st   = (long long)NXc * NYc;                 // time stride
    const long long base = ((long long)ib * Tc + ti) * st;

    const bool active = (y >= SKIPc) && (y <= NYc - 1 - SKIPc);  // y in [15,112]

    // precomputed stencil scale factors
    const float c_dx1 = 1.0f / (12.0f * (4.0f / NXc));                   // 1/(12 DX)
    const float c_dx2 = 1.0f / (12.0f * (4.0f/NXc) * (4.0f/NXc));       // 1/(12 DX^2)
    const float c_dy1 = (NYc - 1) * 0.5f;                                // 1/(2 DY)
    const float c_dy2 = (float)((NYc - 1) * (NYc - 1));                  // 1/DY^2
    const float c_dt  = 2.0f;                                            // 1/(2*0.25)

    float s = 0.0f;

    #pragma unroll
    for (int xi = 0; xi < XPB; ++xi) {
        if (active) {
            const int x   = blockIdx.x * XPB + xi;
            const int xm1 = (x + NXc - 1) & (NXc - 1);   // periodic in x
            const int xp1 = (x + 1)       & (NXc - 1);
            const int xm2 = (x + NXc - 2) & (NXc - 1);
            const int xp2 = (x + 2)       & (NXc - 1);

            const long long i_c  = base + (long long)x   * NYc + y;
            const long long i_p1 = base + (long long)xp1 * NYc + y;
            const long long i_m1 = base + (long long)xm1 * NYc + y;
            const long long i_p2 = base + (long long)xp2 * NYc + y;
            const long long i_m2 = base + (long long)xm2 * NYc + y;

            // ---- continuity residual ----
            const float u_p1 = u[i_p1], u_m1 = u[i_m1];
            const float u_p2 = u[i_p2], u_m2 = u[i_m2];
            const float u_c  = u[i_c];
            const float v_c  = v[i_c];
            const float v_yp = v[i_c + 1], v_ym = v[i_c - 1];

            const float dx4u = (u_m2 - u_p2 + 8.0f * (u_p1 - u_m1)) * c_dx1;
            const float dygv = (v_yp - v_ym) * c_dy1;
            const float r_cont = dx4u + dygv;

            // ---- buoyancy residual ----
            const float b_c  = bb_[i_c];
            const float b_p1 = bb_[i_p1], b_m1 = bb_[i_m1];
            const float b_p2 = bb_[i_p2], b_m2 = bb_[i_m2];
            const float b_yp = bb_[i_c + 1], b_ym = bb_[i_c - 1];
            const float b_tp = bb_[i_c + st], b_tm = bb_[i_c - st];

            const float dbdt = (b_tp - b_tm) * c_dt;
            const float dx4b = (b_m2 - b_p2 + 8.0f * (b_p1 - b_m1)) * c_dx1;
            const float dygb = (b_yp - b_ym) * c_dy1;
            const float d2xb = (16.0f * (b_p1 + b_m1) - (b_p2 + b_m2) - 30.0f * b_c) * c_dx2;
            const float d2yb = (b_yp - 2.0f * b_c + b_ym) * c_dy2;

            const float r_buoy = dbdt + u_c * dx4b + v_c * dygb - kap * (d2xb + d2yb);

            s += r_cont * r_cont + r_buoy * r_buoy;
        }
    }

    // reconverged here: EXEC all ones, safe for WMMA
    const float wsum = wave_reduce_wmma(s);

    __shared__ float red[4];
    const int lane = threadIdx.x & 31;
    const int wid  = threadIdx.x >> 5;
    if (lane == 0) red[wid] = wsum;
    __syncthreads();
    if (threadIdx.x == 0) {
        partials[(long long)blockIdx.y * gridDim.x + blockIdx.x] =
            red[0] + red[1] + red[2] + red[3];
    }
}

// Deterministic fixed-order final reduction (fp64 accumulate), scale by 1/N.
__global__ void __launch_bounds__(256)
rb_final_reduce(const float* __restrict__ partials, int n, float* __restrict__ out)
{
    __shared__ double sh[256];
    double acc = 0.0;
    for (int i = threadIdx.x; i < n; i += 256) acc += (double)partials[i];
    sh[threadIdx.x] = acc;
    __syncthreads();
    for (int off = 128; off > 0; off >>= 1) {
        if ((int)threadIdx.x < off) sh[threadIdx.x] += sh[threadIdx.x + off];
        __syncthreads();
    }
    if (threadIdx.x == 0) {
        const double invN =
            1.0 / (double)((long long)Bc * TINT * NXc * (NYc - 2 * SKIPc));
        out[0] = (float)(sh[0] * invN);
    }
}

extern "C" void kernel_launch(void* const* d_in, const int* in_sizes, int n_in,
                              void* d_out, int out_size, void* d_ws, size_t ws_size,
                              hipStream_t stream) {
    const float* u     = (const float*)d_in[0];
    const float* v     = (const float*)d_in[1];
    const float* b     = (const float*)d_in[2];
    const float* kappa = (const float*)d_in[3];
    float* out         = (float*)d_out;
    float* partials    = (float*)d_ws;      // 15360 floats = 60 KB scratch

    dim3 grid(NXc / XPB, Bc * TINT);        // (64, 240)
    rb_loss_kernel<<<grid, 128, 0, stream>>>(u, v, b, kappa, partials);

    const int nPart = (NXc / XPB) * (Bc * TINT);
    rb_final_reduce<<<1, 256, 0, stream>>>(partials, nPart, out);
}